// Diffusion_56873956934114
// MI455X (gfx1250) — compile-verified
//
#include <hip/hip_runtime.h>
#include <cstdint>

#define N   192
#define N3  (N*N*N)
#define INV2DX 96.0f          // 1/(2*DX), DX = 1/192
#define MU_REF 1.8e-5f
#define CP_OVER_PR (1005.0f / 0.72f)

// Tile geometry: inner 8x8x16, halo 2 for u/T, halo 1 for tau/q
#define IXT 8
#define IYT 8
#define IZT 16
#define LXT (IXT+4)   // 12
#define LYT (IYT+4)   // 12
#define LZT (IZT+4)   // 20 valid z entries per row
#define LZP 24        // padded row stride (floats): data at pz = lz+2, 96B row pitch
#define MXT (IXT+2)   // 10
#define MYT (IYT+2)   // 10
#define MZT (IZT+2)   // 18
#define MP  (MXT*MYT*MZT)   // 1800 tau points
#define IP  (IXT*IYT*IZT)   // 1024 output points
#define NROWS (LXT*LYT)     // 144 (lx,ly) rows in the load region
#define NTHREADS 256

// One (lx,ly) row of one field: 20 contiguous-in-z elements loaded as
// b64 | b128 b128 b128 b128 | b64. The z-halo is +-2 around a 16-wide tile,
// so even when a chunk wraps the periodic boundary it stays contiguous in
// memory (wrap only occurs at a chunk boundary). LDS row is padded to 24
// floats so the b128 chunks land on 16B-aligned LDS addresses.
__device__ __forceinline__
void async_load_rows(const float* __restrict__ p, uint32_t ldsBase,
                     int tid, int x0, int y0,
                     uint32_t gzA4, uint32_t gzM4, uint32_t gzB4)
{
    for (int r = tid; r < NROWS; r += NTHREADS) {
        int ly = r % LYT;
        int lx = r / LYT;
        int gx = x0 + lx; gx += (gx < 0) ? N : 0; gx -= (gx >= N) ? N : 0;
        int gy = y0 + ly; gy += (gy < 0) ? N : 0; gy -= (gy >= N) ? N : 0;
        uint32_t rowg = (uint32_t)((gx * N + gy) * N) * 4u;   // row base, bytes
        uint32_t lb   = ldsBase + (uint32_t)(r * LZP) * 4u;   // LDS row base (96B mult)

        uint32_t gMid = rowg + gzM4;     // 16B aligned: z = zt*16 + {0,4,8,12}
        // leading b64: lz=0,1 -> pz=2 -> LDS +8 ; global z = (zt*16-2) mod N
        asm volatile("global_load_async_to_lds_b64 %0, %1, %2 offset:0"
                     :: "v"(lb + 8u),  "v"(rowg + gzA4), "s"(p) : "memory");
        // four b128: lz=2..17 -> pz=4,8,12,16 -> LDS +16,+32,+48,+64
        asm volatile("global_load_async_to_lds_b128 %0, %1, %2 offset:0"
                     :: "v"(lb + 16u), "v"(gMid), "s"(p) : "memory");
        asm volatile("global_load_async_to_lds_b128 %0, %1, %2 offset:16"
                     :: "v"(lb + 32u), "v"(gMid), "s"(p) : "memory");
        asm volatile("global_load_async_to_lds_b128 %0, %1, %2 offset:32"
                     :: "v"(lb + 48u), "v"(gMid), "s"(p) : "memory");
        asm volatile("global_load_async_to_lds_b128 %0, %1, %2 offset:48"
                     :: "v"(lb + 64u), "v"(gMid), "s"(p) : "memory");
        // trailing b64: lz=18,19 -> pz=20 -> LDS +80 ; global z = (zt*16+16) mod N
        asm volatile("global_load_async_to_lds_b64 %0, %1, %2 offset:0"
                     :: "v"(lb + 80u), "v"(rowg + gzB4), "s"(p) : "memory");
    }
}

__global__ __launch_bounds__(NTHREADS)
void Diffusion_visc_rhs_kernel(const float* __restrict__ u,
                               const float* __restrict__ T,
                               float* __restrict__ out)
{
    __shared__ float sU[3][LXT][LYT][LZP];
    __shared__ float sT[LXT][LYT][LZP];
    __shared__ float sTau[6][MXT][MYT][MZT];   // 00,11,22,01,02,12
    __shared__ float sQ[3][MXT][MYT][MZT];

    const int tid = threadIdx.x;
    const int zt = blockIdx.x, yt = blockIdx.y, xt = blockIdx.z;
    const int x0 = xt*IXT - 2, y0 = yt*IYT - 2;

    // z chunk start indices (uniform per block), with periodic wrap
    int gzA = zt*IZT - 2;  if (gzA < 0)  gzA += N;   // leading pair
    int gzM = zt*IZT;                                 // middle 16, never wraps
    int gzB = zt*IZT + 16; if (gzB >= N) gzB -= N;   // trailing pair
    const uint32_t gzA4 = (uint32_t)gzA * 4u;
    const uint32_t gzM4 = (uint32_t)gzM * 4u;
    const uint32_t gzB4 = (uint32_t)gzB * 4u;

    const uint32_t ldsU = (uint32_t)(uintptr_t)(&sU[0][0][0][0]);
    const uint32_t ldsT = (uint32_t)(uintptr_t)(&sT[0][0][0]);
    const uint32_t fStride = (uint32_t)(NROWS * LZP) * 4u;   // one field's LDS bytes

    // ---- Stage 1: wide async copies of the halo-2 tile into LDS ----
    async_load_rows(u,        ldsU,             tid, x0, y0, gzA4, gzM4, gzB4);
    async_load_rows(u + N3,   ldsU + fStride,   tid, x0, y0, gzA4, gzM4, gzB4);
    async_load_rows(u + 2*N3, ldsU + 2*fStride, tid, x0, y0, gzA4, gzM4, gzB4);
    async_load_rows(T,        ldsT,             tid, x0, y0, gzA4, gzM4, gzB4);
    asm volatile("s_wait_asynccnt 0" ::: "memory");
    __syncthreads();

    // ---- Stage 2: tau + q on the halo-1 region, all from LDS ----
    for (int l = tid; l < MP; l += NTHREADS) {
        int mz = l % MZT;
        int t  = l / MZT;
        int my = t % MYT;
        int mx = t / MYT;
        int x = mx + 1, y = my + 1, z = mz + 3;   // z includes +2 row padding

        float uc0 = sU[0][x][y][z], uc1 = sU[1][x][y][z], uc2 = sU[2][x][y][z];
        float Tc  = sT[x][y][z];
        // mu = MU_REF * (T*T_REF/T_REF)^0.7 = MU_REF * T^0.7 ; T in (0.5,1.5) > 0
        float mu  = MU_REF * __expf(0.7f * __logf(Tc));
        float kc  = mu * CP_OVER_PR;

        float g00 = (sU[0][x+1][y][z] - sU[0][x-1][y][z]) * INV2DX;
        float g01 = (sU[0][x][y+1][z] - sU[0][x][y-1][z]) * INV2DX;
        float g02 = (sU[0][x][y][z+1] - sU[0][x][y][z-1]) * INV2DX;
        float g10 = (sU[1][x+1][y][z] - sU[1][x-1][y][z]) * INV2DX;
        float g11 = (sU[1][x][y+1][z] - sU[1][x][y-1][z]) * INV2DX;
        float g12 = (sU[1][x][y][z+1] - sU[1][x][y][z-1]) * INV2DX;
        float g20 = (sU[2][x+1][y][z] - sU[2][x-1][y][z]) * INV2DX;
        float g21 = (sU[2][x][y+1][z] - sU[2][x][y-1][z]) * INV2DX;
        float g22 = (sU[2][x][y][z+1] - sU[2][x][y][z-1]) * INV2DX;

        float divu = g00 + g11 + g22;
        float lam  = -(2.0f/3.0f) * mu * divu;
        float t00 = 2.0f*mu*g00 + lam;
        float t11 = 2.0f*mu*g11 + lam;
        float t22 = 2.0f*mu*g22 + lam;
        float t01 = mu*(g01 + g10);
        float t02 = mu*(g02 + g20);
        float t12 = mu*(g12 + g21);

        float dT0 = (sT[x+1][y][z] - sT[x-1][y][z]) * INV2DX;
        float dT1 = (sT[x][y+1][z] - sT[x][y-1][z]) * INV2DX;
        float dT2 = (sT[x][y][z+1] - sT[x][y][z-1]) * INV2DX;

        sTau[0][mx][my][mz] = t00;
        sTau[1][mx][my][mz] = t11;
        sTau[2][mx][my][mz] = t22;
        sTau[3][mx][my][mz] = t01;
        sTau[4][mx][my][mz] = t02;
        sTau[5][mx][my][mz] = t12;
        // q_j = sum_i u_i tau_ij + k dT/dx_j ; energy = div(q)
        sQ[0][mx][my][mz] = uc0*t00 + uc1*t01 + uc2*t02 + kc*dT0;
        sQ[1][mx][my][mz] = uc0*t01 + uc1*t11 + uc2*t12 + kc*dT1;
        sQ[2][mx][my][mz] = uc0*t02 + uc1*t12 + uc2*t22 + kc*dT2;
    }
    __syncthreads();

    // ---- Stage 3: divergences, write [0, mom0..2, energy] ----
    for (int l = tid; l < IP; l += NTHREADS) {
        int iz = l % IZT;
        int t  = l / IZT;
        int iy = t % IYT;
        int ix = t / IYT;
        int x = ix + 1, y = iy + 1, z = iz + 1;   // coords in tau region

        float mom0 = (sTau[0][x+1][y][z] - sTau[0][x-1][y][z]
                    + sTau[3][x][y+1][z] - sTau[3][x][y-1][z]
                    + sTau[4][x][y][z+1] - sTau[4][x][y][z-1]) * INV2DX;
        float mom1 = (sTau[3][x+1][y][z] - sTau[3][x-1][y][z]
                    + sTau[1][x][y+1][z] - sTau[1][x][y-1][z]
                    + sTau[5][x][y][z+1] - sTau[5][x][y][z-1]) * INV2DX;
        float mom2 = (sTau[4][x+1][y][z] - sTau[4][x-1][y][z]
                    + sTau[5][x][y+1][z] - sTau[5][x][y-1][z]
                    + sTau[2][x][y][z+1] - sTau[2][x][y][z-1]) * INV2DX;
        float ener = (sQ[0][x+1][y][z] - sQ[0][x-1][y][z]
                    + sQ[1][x][y+1][z] - sQ[1][x][y-1][z]
                    + sQ[2][x][y][z+1] - sQ[2][x][y][z-1]) * INV2DX;

        int gx = xt*IXT + ix, gy = yt*IYT + iy, gz = zt*IZT + iz;
        size_t idx = (size_t)((gx * N) + gy) * N + gz;
        out[idx]                = 0.0f;   // mass
        out[(size_t)N3   + idx] = mom0;
        out[(size_t)2*N3 + idx] = mom1;
        out[(size_t)3*N3 + idx] = mom2;
        out[(size_t)4*N3 + idx] = ener;
    }
}

extern "C" void kernel_launch(void* const* d_in, const int* in_sizes, int n_in,
                              void* d_out, int out_size, void* d_ws, size_t ws_size,
                              hipStream_t stream) {
    const float* u = (const float*)d_in[0];   // [3,192,192,192]
    const float* T = (const float*)d_in[1];   // [192,192,192]
    float* out = (float*)d_out;               // [5,192,192,192]
    dim3 grid(N / IZT, N / IYT, N / IXT);     // (12, 24, 24)
    Diffusion_visc_rhs_kernel<<<grid, NTHREADS, 0, stream>>>(u, T, out);
}